// GCNEncoder_24206435681065
// MI455X (gfx1250) — compile-verified
//
#include <hip/hip_runtime.h>
#include <hip/hip_fp16.h>

// ---------------------------------------------------------------------------
// GAT (2 layers, heads=1) for MI455X / gfx1250.
//   layer: h = X@W ; a_s = h.a_src ; a_d = h.a_dst
//          e = leaky_relu(a_s[src]+a_d[dst]) ; segment softmax over dst
//          out[dst] += h[src]*alpha ; + bias ; relu (L1) / sigmoid (L2)
// GEMMs run on the fp32 matrix pipe (v_wmma_f32_16x16x4_f32) with both A and
// B tiles staged in LDS (ds_load_2addr_b64 feeds the fragments). The edge
// phase is L2-resident: vectorized gather of h[src,:] + fp32 atomic scatter
// (global_atomic_add_f32); exact segment-max via order-preserving int keys.
// ---------------------------------------------------------------------------

typedef __attribute__((ext_vector_type(2))) float v2f;
typedef __attribute__((ext_vector_type(8))) float v8f;

// ----- order-preserving float<->int key (for exact atomic float max) -------
__device__ __forceinline__ int f2ord(float f) {
  int i = __float_as_int(f);
  return (i >= 0) ? i : (i ^ 0x7FFFFFFF);
}
__device__ __forceinline__ float ord2f(int k) {
  return __int_as_float((k >= 0) ? k : (k ^ 0x7FFFFFFF));
}
#define NEG_INF_ORD 0x807FFFFF   // f2ord(-inf)

// ---------------------------------------------------------------------------
// WMMA GEMM: H[N,FOUT] = X[N,K] @ W[K,FOUT].  One 16x16 output tile per wave;
// FOUT/16 waves per block. X row-tile and transposed W staged in LDS so the
// inner loop is pure ds_load + v_wmma_f32_16x16x4_f32.
// Requires N%16==0, FOUT%16==0, K%4==0 (here N=50000, K=128, FOUT=128/64).
// ---------------------------------------------------------------------------
template <int K, int FOUT>
__global__ void wmma_gemm_kernel(const float* __restrict__ X,
                                 const float* __restrict__ W,
                                 float* __restrict__ Hout) {
  __shared__ float Xs[16][K + 4];      // +4 pad: dodge LDS bank conflicts
  __shared__ float Ws[FOUT][K + 4];    // W transposed: Ws[col][k]
  const int rowBase = blockIdx.x * 16;
  const int wave = threadIdx.x >> 5;
  const int lane = threadIdx.x & 31;
  const int hi   = lane >> 4;          // half-wave select
  const int lr   = lane & 15;

  for (int i = threadIdx.x; i < 16 * K; i += blockDim.x) {
    int r = i / K, c = i % K;
    Xs[r][c] = X[(size_t)(rowBase + r) * K + c];
  }
  for (int i = threadIdx.x; i < K * FOUT; i += blockDim.x) {
    int k = i / FOUT, c = i % FOUT;    // W[i] is coalesced; LDS write scattered
    Ws[c][k] = W[i];
  }
  __syncthreads();

  const int colBase = wave * 16;       // (blockDim/32)*16 == FOUT
  v8f acc = {};

#pragma unroll
  for (int k0 = 0; k0 < K; k0 += 4) {
    v2f a, b;
    // A 16x4: lanes 0-15 hold K=k0+0/1, lanes 16-31 K=k0+2/3, row M=lr
    a.x = Xs[lr][k0 + 2 * hi + 0];
    a.y = Xs[lr][k0 + 2 * hi + 1];
    // B 4x16 (mirror): col N=lr, K split across half-waves
    b.x = Ws[colBase + lr][k0 + 2 * hi + 0];
    b.y = Ws[colBase + lr][k0 + 2 * hi + 1];
    acc = __builtin_amdgcn_wmma_f32_16x16x4_f32(
        false, a, false, b, (short)0, acc, false, false);
  }

  // D layout: VGPR r -> row rowBase + 8*hi + r, col colBase + lr
#pragma unroll
  for (int r = 0; r < 8; ++r)
    Hout[(size_t)(rowBase + 8 * hi + r) * FOUT + colBase + lr] = acc[r];
}

// ---------------------------------------------------------------------------
// Per-node attention logits: as[n] = h[n,:].a_src, ad[n] = h[n,:].a_dst
// One wave32 per node, shuffle reduction.
// ---------------------------------------------------------------------------
__global__ void alpha_kernel(const float* __restrict__ Hf,
                             const float* __restrict__ a_src,
                             const float* __restrict__ a_dst,
                             float* __restrict__ as, float* __restrict__ ad,
                             int N, int F) {
  int node = (blockIdx.x * blockDim.x + threadIdx.x) >> 5;
  int lane = threadIdx.x & 31;
  if (node >= N) return;
  float ss = 0.f, sd = 0.f;
  for (int f = lane; f < F; f += 32) {
    float hv = Hf[(size_t)node * F + f];
    ss += hv * a_src[f];
    sd += hv * a_dst[f];
  }
#pragma unroll
  for (int off = 16; off > 0; off >>= 1) {
    ss += __shfl_xor(ss, off, 32);
    sd += __shfl_xor(sd, off, 32);
  }
  if (lane == 0) { as[node] = ss; ad[node] = sd; }
}

__global__ void seg_init_kernel(int* __restrict__ mkey,
                                float* __restrict__ denom, int N) {
  int i = blockIdx.x * blockDim.x + threadIdx.x;
  if (i < N) { mkey[i] = NEG_INF_ORD; denom[i] = 0.f; }
}

__device__ __forceinline__ void edge_sd(const int* __restrict__ ei, int E,
                                        int idx, int& s, int& d) {
  if (idx < E) { s = ei[idx]; d = ei[(size_t)E + idx]; }
  else         { s = d = idx - E; }          // self loop
}

__global__ void edge_max_kernel(const int* __restrict__ ei, int E, int N,
                                const float* __restrict__ as,
                                const float* __restrict__ ad,
                                int* __restrict__ mkey) {
  int idx = blockIdx.x * blockDim.x + threadIdx.x;
  if (idx >= E + N) return;
  int s, d; edge_sd(ei, E, idx, s, d);
  float e = as[s] + ad[d];
  e = (e > 0.f) ? e : 0.2f * e;              // leaky_relu(0.2)
  atomicMax(&mkey[d], f2ord(e));
}

__global__ void edge_expsum_kernel(const int* __restrict__ ei, int E, int N,
                                   const float* __restrict__ as,
                                   const float* __restrict__ ad,
                                   const int* __restrict__ mkey,
                                   float* __restrict__ exbuf,
                                   float* __restrict__ denom) {
  int idx = blockIdx.x * blockDim.x + threadIdx.x;
  if (idx >= E + N) return;
  int s, d; edge_sd(ei, E, idx, s, d);
  float e = as[s] + ad[d];
  e = (e > 0.f) ? e : 0.2f * e;
  float ex = __expf(e - ord2f(mkey[d]));
  exbuf[idx] = ex;
  atomicAdd(&denom[d], ex);
}

// One wave32 per edge: gather h[src,:] (contiguous F/32-wide chunk per lane,
// vectorizes to global_load_b128/b64), scale by alpha, scatter-add to dst.
template <int F>
__global__ void edge_aggregate_kernel(const int* __restrict__ ei, int E, int N,
                                      const float* __restrict__ Hf,
                                      const float* __restrict__ exbuf,
                                      const float* __restrict__ denom,
                                      float* __restrict__ outacc) {
  constexpr int VW = F / 32;
  int idx = (blockIdx.x * blockDim.x + threadIdx.x) >> 5;
  int lane = threadIdx.x & 31;
  if (idx >= E + N) return;
  int s, d; edge_sd(ei, E, idx, s, d);
  const float* hrow = Hf + (size_t)s * F + lane * VW;
  float* orow = outacc + (size_t)d * F + lane * VW;
  __builtin_prefetch(hrow, 0, 1);            // global_prefetch_b8
  float v[VW];
#pragma unroll
  for (int j = 0; j < VW; ++j) v[j] = hrow[j];
  float coef = exbuf[idx] / denom[d];
#pragma unroll
  for (int j = 0; j < VW; ++j) atomicAdd(&orow[j], v[j] * coef);
}

__global__ void finalize_relu_kernel(float* __restrict__ acc,
                                     const float* __restrict__ bias,
                                     int total, int fmask) {
  int i = blockIdx.x * blockDim.x + threadIdx.x;
  if (i >= total) return;
  float v = acc[i] + bias[i & fmask];
  acc[i] = v > 0.f ? v : 0.f;
}

__global__ void finalize_sigmoid_kernel(const float* __restrict__ acc,
                                        const float* __restrict__ bias,
                                        float* __restrict__ out,
                                        int total, int fmask) {
  int i = blockIdx.x * blockDim.x + threadIdx.x;
  if (i >= total) return;
  float v = acc[i] + bias[i & fmask];
  out[i] = 1.f / (1.f + __expf(-v));
}

// ---------------------------------------------------------------------------
static inline int cdiv(long long a, long long b) { return (int)((a + b - 1) / b); }

extern "C" void kernel_launch(void* const* d_in, const int* in_sizes, int n_in,
                              void* d_out, int out_size, void* d_ws, size_t ws_size,
                              hipStream_t stream) {
  const float* x    = (const float*)d_in[0];
  const int*   ei   = (const int*)d_in[1];   // [2,E] int32 (row0=src,row1=dst)
  const float* W1   = (const float*)d_in[2];
  const float* a1s  = (const float*)d_in[3];
  const float* a1d  = (const float*)d_in[4];
  const float* b1   = (const float*)d_in[5];
  const float* W2   = (const float*)d_in[6];
  const float* a2s  = (const float*)d_in[7];
  const float* a2d  = (const float*)d_in[8];
  const float* b2   = (const float*)d_in[9];
  float* out        = (float*)d_out;

  const int H   = in_sizes[3];              // 128
  const int Fin = in_sizes[2] / H;          // 128
  const int Fo  = in_sizes[7];              // 64
  const int N   = in_sizes[0] / Fin;        // 50000
  const int E   = in_sizes[1] / 2;          // 800000
  const int EN  = E + N;

  // ---- workspace carving (256B aligned) ----
  char* w = (char*)d_ws;
  size_t off = 0;
  auto carve = [&](size_t bytes) {
    void* p = w + off;
    off += (bytes + 255) & ~(size_t)255;
    return p;
  };
  float* h1    = (float*)carve((size_t)N * H  * 4);
  float* acc1  = (float*)carve((size_t)N * H  * 4);
  float* h2    = (float*)carve((size_t)N * Fo * 4);
  float* acc2  = (float*)carve((size_t)N * Fo * 4);
  float* as1   = (float*)carve((size_t)N * 4);
  float* ad1   = (float*)carve((size_t)N * 4);
  float* as2   = (float*)carve((size_t)N * 4);
  float* ad2   = (float*)carve((size_t)N * 4);
  int*   mkey  = (int*)  carve((size_t)N * 4);
  float* denom = (float*)carve((size_t)N * 4);
  float* exbuf = (float*)carve((size_t)EN * 4);
  (void)ws_size; (void)n_in; (void)out_size;

  const int TB = 256;

  // ================= layer 1 =================
  hipMemsetAsync(acc1, 0, (size_t)N * H * 4, stream);
  wmma_gemm_kernel<128, 128><<<N / 16, 8 * 32, 0, stream>>>(x, W1, h1);
  alpha_kernel<<<cdiv((long long)N * 32, TB), TB, 0, stream>>>(h1, a1s, a1d,
                                                               as1, ad1, N, H);
  seg_init_kernel<<<cdiv(N, TB), TB, 0, stream>>>(mkey, denom, N);
  edge_max_kernel<<<cdiv(EN, TB), TB, 0, stream>>>(ei, E, N, as1, ad1, mkey);
  edge_expsum_kernel<<<cdiv(EN, TB), TB, 0, stream>>>(ei, E, N, as1, ad1,
                                                      mkey, exbuf, denom);
  edge_aggregate_kernel<128><<<cdiv((long long)EN * 32, TB), TB, 0, stream>>>(
      ei, E, N, h1, exbuf, denom, acc1);
  finalize_relu_kernel<<<cdiv((long long)N * H, TB), TB, 0, stream>>>(
      acc1, b1, N * H, H - 1);

  // ================= layer 2 =================
  hipMemsetAsync(acc2, 0, (size_t)N * Fo * 4, stream);
  wmma_gemm_kernel<128, 64><<<N / 16, 4 * 32, 0, stream>>>(acc1, W2, h2);
  alpha_kernel<<<cdiv((long long)N * 32, TB), TB, 0, stream>>>(h2, a2s, a2d,
                                                               as2, ad2, N, Fo);
  seg_init_kernel<<<cdiv(N, TB), TB, 0, stream>>>(mkey, denom, N);
  edge_max_kernel<<<cdiv(EN, TB), TB, 0, stream>>>(ei, E, N, as2, ad2, mkey);
  edge_expsum_kernel<<<cdiv(EN, TB), TB, 0, stream>>>(ei, E, N, as2, ad2,
                                                      mkey, exbuf, denom);
  edge_aggregate_kernel<64><<<cdiv((long long)EN * 32, TB), TB, 0, stream>>>(
      ei, E, N, h2, exbuf, denom, acc2);
  finalize_sigmoid_kernel<<<cdiv((long long)N * Fo, TB), TB, 0, stream>>>(
      acc2, b2, out, N * Fo, Fo - 1);
}